// unfrozen_DyanC_10608569221405
// MI455X (gfx1250) — compile-verified
//
#include <hip/hip_runtime.h>
#include <math.h>

// ---------------------------------------------------------------------------
// FISTA sparse coding on MI455X (gfx1250).
// Compute-bound: 212 GFLOP fp32 vs ~60MB HBM traffic -> keep A (124KB) and
// all iteration state resident in LDS/VGPRs; fp32 WMMA (16x16x4) for exact
// reference semantics. TDM (tensor_load_to_lds) stages A and D^T into LDS.
// ---------------------------------------------------------------------------

#define T_DIM   36
#define N_AT    40
#define K_DIM   161          // 1 + 4*40
#define B_DIM   2
#define F_DIM   20480
#define KP      176          // K padded to 11*16 (zero-padded rows/cols)
#define AST     180          // LDS/global row stride for A (banks: gcd(52,64)=4 -> conflict-free frags)
#define DTST    44           // row stride for D^T
#define NWAVE   8
#define TILE_F  (16*NWAVE)   // 128 columns per block, 16 per wave
#define MT      (KP/16)      // 11 M-tiles
#define KK_A    (KP/4)       // 44 K-steps for A@y
#define KK_D    (T_DIM/4)    // 9  K-steps for Dt@Y
#define MAX_ITER 100
#define LAMBDA  0.1f

// workspace layout (in floats)
#define WS_A    16
#define WS_DT   (WS_A + KP*AST)

typedef float v2f  __attribute__((ext_vector_type(2)));
typedef float v8f  __attribute__((ext_vector_type(8)));
typedef unsigned int u32x4 __attribute__((ext_vector_type(4)));
typedef int   i32x8 __attribute__((ext_vector_type(8)));
typedef int   i32x4 __attribute__((ext_vector_type(4)));

static __device__ inline v8f v8zero() {
    v8f r = {0.f,0.f,0.f,0.f,0.f,0.f,0.f,0.f};
    return r;
}

// ---------------------------------------------------------------------------
// Stage 1 (one block): build normalized dictionary D [36x161], DtD, Frobenius
// norm -> linv, A = I - DtD*linv padded to [176x180], D^T padded to [176x44].
// ---------------------------------------------------------------------------
__global__ __launch_bounds__(256) void fista_prep(const float* __restrict__ Drr,
                                                  const float* __restrict__ Dth,
                                                  float* __restrict__ ws) {
    __shared__ float sD[T_DIM * K_DIM];     // D row-major [t][k]
    __shared__ float sDtD[K_DIM * K_DIM];
    __shared__ float red[256];
    const int tid = threadIdx.x;

    if (tid < K_DIM) {
        const int k = tid;
        float col[T_DIM];
        if (k == 0) {
            for (int t = 0; t < T_DIM; ++t) col[t] = 1.f;
        } else {
            const int g = (k - 1) / N_AT;
            const int j = (k - 1) % N_AT;
            const float r  = Drr[j];
            const float th = Dth[j];
            float ri = 1.f;                     // r^t
            for (int t = 0; t < T_DIM; ++t) {
                const float c = cosf((float)t * th);
                const float s = sinf((float)t * th);
                const float sgn = (t & 1) ? -1.f : 1.f;
                float v;
                if      (g == 0) v = ri * c;
                else if (g == 1) v = sgn * ri * c;
                else if (g == 2) v = ri * s;
                else             v = sgn * ri * s;
                col[t] = v;
                ri *= r;
            }
        }
        float nrm = 0.f;
        for (int t = 0; t < T_DIM; ++t) nrm += col[t] * col[t];
        nrm = sqrtf(nrm);
        if (nrm == 0.f) nrm = sqrtf((float)T_DIM);
        const float inv = 1.f / nrm;
        for (int t = 0; t < T_DIM; ++t) sD[t * K_DIM + k] = col[t] * inv;
    }
    __syncthreads();

    // DtD and partial Frobenius sum
    float fsum = 0.f;
    for (int idx = tid; idx < K_DIM * K_DIM; idx += 256) {
        const int i = idx / K_DIM, j = idx % K_DIM;
        float d = 0.f;
        for (int t = 0; t < T_DIM; ++t) d += sD[t * K_DIM + i] * sD[t * K_DIM + j];
        sDtD[idx] = d;
        fsum += d * d;
    }
    red[tid] = fsum;
    __syncthreads();
    for (int s = 128; s > 0; s >>= 1) {
        if (tid < s) red[tid] += red[tid + s];
        __syncthreads();
    }
    const float linv = 1.f / sqrtf(red[0]);
    if (tid == 0) { ws[0] = linv; ws[1] = LAMBDA * linv; }

    // A padded [KP x AST]
    for (int idx = tid; idx < KP * AST; idx += 256) {
        const int i = idx / AST, j = idx % AST;
        float v = 0.f;
        if (i < K_DIM && j < K_DIM)
            v = ((i == j) ? 1.f : 0.f) - sDtD[i * K_DIM + j] * linv;
        ws[WS_A + idx] = v;
    }
    // D^T padded [KP x DTST]
    for (int idx = tid; idx < KP * DTST; idx += 256) {
        const int k = idx / DTST, t = idx % DTST;
        ws[WS_DT + idx] = (k < K_DIM && t < T_DIM) ? sD[t * K_DIM + k] : 0.f;
    }
}

// ---------------------------------------------------------------------------
// Stage 2: persistent-tile FISTA. Each wave owns 16 columns; A and D^T shared
// in LDS (TDM-loaded); y-state in per-wave LDS stripe; DtY/x_old in WMMA regs.
// ---------------------------------------------------------------------------
__global__ __launch_bounds__(256) void fista_iter(const float* __restrict__ Y,
                                                  const float* __restrict__ ws,
                                                  float* __restrict__ out) {
    __shared__ float sA [KP * AST];        // 126720 B
    __shared__ float sDt[KP * DTST];       //  30976 B
    __shared__ float sY [TILE_F * AST];    //  92160 B  (col-major: [col][row])

    const int tid  = threadIdx.x;
    const int lane = tid & 31;
    const int w    = tid >> 5;
    const int half = lane >> 4;            // 0: lanes 0-15, 1: lanes 16-31
    const int l16  = lane & 15;
    const int wc0  = w * 16;               // wave's column base within tile

    const int blocksPerB = F_DIM / TILE_F; // 160
    const int b  = blockIdx.x / blocksPerB;
    const int f0 = (blockIdx.x % blocksPerB) * TILE_F;

#if __has_builtin(__builtin_amdgcn_tensor_load_to_lds) && __has_builtin(__builtin_amdgcn_s_wait_tensorcnt)
    if (tid < 32) {  // one wave drives the TDM; others sync at the barrier
        const unsigned long long gaA = (unsigned long long)(const void*)(ws + WS_A);
        const unsigned int ldsA = (unsigned int)(unsigned long long)(void*)&sA[0];
        // D# group0: count=1 | lds_addr | global_addr | type=2
        u32x4 g0 = { 1u, ldsA, (unsigned)gaA,
                     (unsigned)(gaA >> 32) | (2u << 30) };
        // D# group1: data_size=4B, 2D tensor dim0=AST dim1=KP, tile = full, stride0=AST
        i32x8 g1 = { (int)(2u << 16),
                     (int)((unsigned)AST << 16),
                     (int)((unsigned)KP  << 16),
                     (int)((unsigned)AST << 16),
                     (int)KP, (int)AST, 0, 0 };
        i32x4 z4 = { 0, 0, 0, 0 };
        i32x8 z8 = { 0, 0, 0, 0, 0, 0, 0, 0 };
        __builtin_amdgcn_tensor_load_to_lds(g0, g1, z4, z4, z8, 0);

        const unsigned long long gaD = (unsigned long long)(const void*)(ws + WS_DT);
        const unsigned int ldsD = (unsigned int)(unsigned long long)(void*)&sDt[0];
        u32x4 h0 = { 1u, ldsD, (unsigned)gaD,
                     (unsigned)(gaD >> 32) | (2u << 30) };
        i32x8 h1 = { (int)(2u << 16),
                     (int)((unsigned)DTST << 16),
                     (int)((unsigned)KP   << 16),
                     (int)((unsigned)DTST << 16),
                     (int)KP, (int)DTST, 0, 0 };
        __builtin_amdgcn_tensor_load_to_lds(h0, h1, z4, z4, z8, 0);
        __builtin_amdgcn_s_wait_tensorcnt(0);
    }
#else
    for (int i = tid; i < KP * AST;  i += 256) sA[i]  = ws[WS_A + i];
    for (int i = tid; i < KP * DTST; i += 256) sDt[i] = ws[WS_DT + i];
#endif

    // Stage Y tile transposed into sY: sY[c][t] = Y[b][t][f0+c]  (coalesced in c)
    for (int i = tid; i < T_DIM * TILE_F; i += 256) {
        const int t = i / TILE_F, c = i % TILE_F;
        sY[c * AST + t] = Y[((size_t)b * T_DIM + t) * F_DIM + f0 + c];
    }
    __syncthreads();

    const float linv = ws[0];
    const float lam  = ws[1];

    // DtY = linv * D^T @ Ytile, via fp32 WMMA (11 M-tiles x 9 K-steps)
    v8f dty[MT];
    const int bcol = (wc0 + l16) * AST;    // this lane's B-fragment column base
    #pragma unroll
    for (int m = 0; m < MT; ++m) {
        v8f acc = v8zero();
        const int arow = (m * 16 + l16) * DTST;
        for (int kk = 0; kk < KK_D; ++kk) {
            const int kc = 4 * kk + 2 * half;
            v2f a  = *(const v2f*)&sDt[arow + kc];
            v2f bb = *(const v2f*)&sY[bcol + kc];
            acc = __builtin_amdgcn_wmma_f32_16x16x4_f32(
                      false, a, false, bb, (short)0, acc, false, false);
        }
        #pragma unroll
        for (int v = 0; v < 8; ++v) acc[v] *= linv;
        dty[m] = acc;
    }
    __syncthreads();

    // y0 = 0, x0 = 0
    for (int i = tid; i < TILE_F * AST; i += 256) sY[i] = 0.f;
    __syncthreads();

    v8f xold[MT];
    #pragma unroll
    for (int m = 0; m < MT; ++m) xold[m] = v8zero();

    // FISTA main loop. Each wave touches only its own sY stripe, and DS ops
    // from one wave are processed in order -> no barrier needed per iteration.
    float ts = 1.f;
    for (int it = 0; it < MAX_ITER; ++it) {
        v8f acc[MT];
        #pragma unroll
        for (int m = 0; m < MT; ++m) acc[m] = dty[m];

        for (int kk = 0; kk < KK_A; ++kk) {
            const int kc = 4 * kk + 2 * half;
            v2f bb = *(const v2f*)&sY[bcol + kc];   // y fragment, reused over 11 M-tiles
            #pragma unroll
            for (int m = 0; m < MT; ++m) {
                v2f a = *(const v2f*)&sA[(m * 16 + l16) * AST + kc];
                acc[m] = __builtin_amdgcn_wmma_f32_16x16x4_f32(
                             false, a, false, bb, (short)0, acc[m], false, false);
            }
        }

        const float tn = 0.5f * (1.f + sqrtf(1.f + 4.f * ts * ts));
        const float tt = (ts - 1.f) / tn;
        ts = tn;
        const float c1 = 1.f + tt;

        #pragma unroll
        for (int m = 0; m < MT; ++m) {
            const int rb = m * 16 + 8 * half;       // D-layout: row = 16m + v + 8*half
            #pragma unroll
            for (int v = 0; v < 8; ++v) {
                const float z  = acc[m][v];
                const float az = fabsf(z) - lam;
                const float xn = (az > 0.f) ? copysignf(az, z) : 0.f;  // softshrink
                const float yn = c1 * xn - tt * xold[m][v];            // momentum
                xold[m][v] = xn;
                sY[bcol + rb + v] = yn;
            }
        }
    }

    // Emit x (rows < 161) coalesced over f
    const int f = f0 + wc0 + l16;
    #pragma unroll
    for (int m = 0; m < MT; ++m) {
        const int rb = m * 16 + 8 * half;
        #pragma unroll
        for (int v = 0; v < 8; ++v) {
            const int row = rb + v;
            if (row < K_DIM)
                out[((size_t)b * K_DIM + row) * F_DIM + f] = xold[m][v];
        }
    }
}

extern "C" void kernel_launch(void* const* d_in, const int* in_sizes, int n_in,
                              void* d_out, int out_size, void* d_ws, size_t ws_size,
                              hipStream_t stream) {
    const float* x   = (const float*)d_in[0];   // [2,36,20480]
    const float* Drr = (const float*)d_in[1];   // [40]
    const float* Dth = (const float*)d_in[2];   // [40]
    float* out = (float*)d_out;                 // [2,161,20480]
    float* ws  = (float*)d_ws;                  // scalars + A_pad + Dt_pad (~158 KB)

    fista_prep<<<1, 256, 0, stream>>>(Drr, Dth, ws);
    fista_iter<<<(B_DIM * F_DIM) / TILE_F, 256, 0, stream>>>(x, ws, out);
}